// TemporalGCN_88897233092609
// MI455X (gfx1250) — compile-verified
//
#include <hip/hip_runtime.h>
#include <hip/hip_bf16.h>
#include <math.h>

typedef __attribute__((ext_vector_type(2))) float v2f;
typedef __attribute__((ext_vector_type(4))) float v4f;
typedef __attribute__((ext_vector_type(8))) float v8f;

#define NN 50000
#define EE 1600000
#define NT (NN / 16)   // 3125 row tiles of 16 nodes

// Hardware f32 atomic add (global_atomic_add_f32, no CAS loop).
__device__ __forceinline__ void atomAddF32(float* p, float v) {
  unsafeAtomicAdd(p, v);
}

// ---------------- utility kernels ----------------

__global__ void k_zero(float* __restrict__ p, int n) {
  int i = blockIdx.x * blockDim.x + threadIdx.x;
  if (i < n) p[i] = 0.0f;
}

__global__ void k_degree(const int* __restrict__ dst, float* __restrict__ deg) {
  int e = blockIdx.x * blockDim.x + threadIdx.x;
  if (e < EE) atomAddF32(&deg[dst[e]], 1.0f);
}

__global__ void k_dinv(const float* __restrict__ deg, float* __restrict__ dinv) {
  int n = blockIdx.x * blockDim.x + threadIdx.x;
  if (n < NN) {
    float d = deg[n];
    dinv[n] = (d > 0.0f) ? rsqrtf(d) : 0.0f;
  }
}

// ---------------- WMMA feature transform: Hz = x@Wz_c, Hh = x@Wh_c ----------------
// One wave handles a 16-row tile; fp32 WMMA 16x16x4, K=32 (8 steps), 2 col tiles x 2 gates.
// A layout (16x4 f32): lane l: row = l&15, khalf = l>>4; VGPR v holds K = 2*khalf + v.
// B layout (4x16 f32): VGPR v, lane l holds B[K = v + 2*(l>>4)][N = l&15].
// D layout (16x16 f32): VGPR r, lane l holds D[M = r + 8*(l>>4)][N = l&15].

__device__ __forceinline__ v8f wmma_f32(v2f a, v2f b, v8f c) {
  return __builtin_amdgcn_wmma_f32_16x16x4_f32(false, a, false, b, (short)0, c, false, false);
}

__global__ void k_feat(const float* __restrict__ x,
                       const float* __restrict__ Wz,
                       const float* __restrict__ Wh,
                       float* __restrict__ Hz,
                       float* __restrict__ Hh) {
  int wave = (blockIdx.x * blockDim.x + threadIdx.x) >> 5;
  if (wave >= NT) return;
  int lane  = threadIdx.x & 31;
  int col   = lane & 15;
  int khalf = lane >> 4;
  int row   = wave * 16 + col;   // A-fragment row for this lane

  v8f cz0 = {}, cz1 = {}, ch0 = {}, ch1 = {};
  #pragma unroll
  for (int k0 = 0; k0 < 32; k0 += 4) {
    int ka = k0 + 2 * khalf;
    v2f a;
    a.x = x[row * 32 + ka + 0];
    a.y = x[row * 32 + ka + 1];
    v2f bz0, bz1, bh0, bh1;
    bz0.x = Wz[(ka + 0) * 32 + col];      bz0.y = Wz[(ka + 1) * 32 + col];
    bz1.x = Wz[(ka + 0) * 32 + 16 + col]; bz1.y = Wz[(ka + 1) * 32 + 16 + col];
    bh0.x = Wh[(ka + 0) * 32 + col];      bh0.y = Wh[(ka + 1) * 32 + col];
    bh1.x = Wh[(ka + 0) * 32 + 16 + col]; bh1.y = Wh[(ka + 1) * 32 + 16 + col];
    cz0 = wmma_f32(a, bz0, cz0);
    cz1 = wmma_f32(a, bz1, cz1);
    ch0 = wmma_f32(a, bh0, ch0);
    ch1 = wmma_f32(a, bh1, ch1);
  }
  #pragma unroll
  for (int r = 0; r < 8; ++r) {
    int m = wave * 16 + r + 8 * khalf;
    Hz[m * 32 + col]      = cz0[r];
    Hz[m * 32 + 16 + col] = cz1[r];
    Hh[m * 32 + col]      = ch0[r];
    Hh[m * 32 + 16 + col] = ch1[r];
  }
}

// ---------------- edge gather/scale/scatter ----------------
// 8 threads per edge, 4 channels each (float4 gather, 4 hw f32 atomics per gate).

__global__ void k_edge(const int* __restrict__ src, const int* __restrict__ dst,
                       const float* __restrict__ dinv,
                       const float* __restrict__ Hz, const float* __restrict__ Hh,
                       float* __restrict__ Az, float* __restrict__ Ah) {
  int tid = blockIdx.x * blockDim.x + threadIdx.x;
  int e  = tid >> 3;
  if (e >= EE) return;
  int cg = (tid & 7) * 4;
  int s = src[e], d = dst[e];
  float norm = dinv[s] * dinv[d];
  v4f hz = *(const v4f*)(Hz + s * 32 + cg);
  v4f hh = *(const v4f*)(Hh + s * 32 + cg);
  float* az = Az + d * 32 + cg;
  float* ah = Ah + d * 32 + cg;
  atomAddF32(az + 0, hz.x * norm);
  atomAddF32(az + 1, hz.y * norm);
  atomAddF32(az + 2, hz.z * norm);
  atomAddF32(az + 3, hz.w * norm);
  atomAddF32(ah + 0, hh.x * norm);
  atomAddF32(ah + 1, hh.y * norm);
  atomAddF32(ah + 2, hh.z * norm);
  atomAddF32(ah + 3, hh.w * norm);
}

// ---------------- WMMA gate GEMMs + fused pointwise epilogue ----------------
// Z  = sigmoid((Az + bz_c) @ Wz_l[:32,:] + bz_l)
// Ht = tanh   ((Ah + bh_c) @ Wh_l[:32,:] + bh_l)
// Hrelu = relu((1 - Z) * Ht)

__global__ void k_gate(const float* __restrict__ Az, const float* __restrict__ Ah,
                       const float* __restrict__ bz_c, const float* __restrict__ bh_c,
                       const float* __restrict__ Wz_l, const float* __restrict__ Wh_l,
                       const float* __restrict__ bz_l, const float* __restrict__ bh_l,
                       float* __restrict__ Hrelu) {
  int wave = (blockIdx.x * blockDim.x + threadIdx.x) >> 5;
  if (wave >= NT) return;
  int lane  = threadIdx.x & 31;
  int col   = lane & 15;
  int khalf = lane >> 4;
  int row   = wave * 16 + col;

  v8f cz0 = {}, cz1 = {}, ch0 = {}, ch1 = {};
  #pragma unroll
  for (int k0 = 0; k0 < 32; k0 += 4) {
    int ka = k0 + 2 * khalf;
    v2f az, ah;
    az.x = Az[row * 32 + ka + 0] + bz_c[ka + 0];
    az.y = Az[row * 32 + ka + 1] + bz_c[ka + 1];
    ah.x = Ah[row * 32 + ka + 0] + bh_c[ka + 0];
    ah.y = Ah[row * 32 + ka + 1] + bh_c[ka + 1];
    v2f bz0, bz1, bh0, bh1;   // Wz_l / Wh_l are [64,32] row-major; rows 0..31 used
    bz0.x = Wz_l[(ka + 0) * 32 + col];      bz0.y = Wz_l[(ka + 1) * 32 + col];
    bz1.x = Wz_l[(ka + 0) * 32 + 16 + col]; bz1.y = Wz_l[(ka + 1) * 32 + 16 + col];
    bh0.x = Wh_l[(ka + 0) * 32 + col];      bh0.y = Wh_l[(ka + 1) * 32 + col];
    bh1.x = Wh_l[(ka + 0) * 32 + 16 + col]; bh1.y = Wh_l[(ka + 1) * 32 + 16 + col];
    cz0 = wmma_f32(az, bz0, cz0);
    cz1 = wmma_f32(az, bz1, cz1);
    ch0 = wmma_f32(ah, bh0, ch0);
    ch1 = wmma_f32(ah, bh1, ch1);
  }
  float bZ0 = bz_l[col], bZ1 = bz_l[16 + col];
  float bH0 = bh_l[col], bH1 = bh_l[16 + col];
  #pragma unroll
  for (int r = 0; r < 8; ++r) {
    int m = wave * 16 + r + 8 * khalf;
    float z  = 1.0f / (1.0f + expf(-(cz0[r] + bZ0)));
    float ht = tanhf(ch0[r] + bH0);
    float hn = (1.0f - z) * ht;
    Hrelu[m * 32 + col] = fmaxf(hn, 0.0f);
    z  = 1.0f / (1.0f + expf(-(cz1[r] + bZ1)));
    ht = tanhf(ch1[r] + bH1);
    hn = (1.0f - z) * ht;
    Hrelu[m * 32 + 16 + col] = fmaxf(hn, 0.0f);
  }
}

// ---------------- readout: out[n] = relu_h[n,:] . W_lin + b_lin ----------------

__global__ void k_out(const float* __restrict__ Hrelu,
                      const float* __restrict__ W_lin,
                      const float* __restrict__ b_lin,
                      float* __restrict__ out) {
  int n = blockIdx.x * blockDim.x + threadIdx.x;
  if (n >= NN) return;
  const v4f* hp = (const v4f*)(Hrelu + n * 32);
  const v4f* wp = (const v4f*)W_lin;
  float acc = b_lin[0];
  #pragma unroll
  for (int i = 0; i < 8; ++i) {
    v4f h = hp[i], w = wp[i];
    acc += h.x * w.x + h.y * w.y + h.z * w.z + h.w * w.w;
  }
  out[n] = acc;
}

// ---------------- launcher ----------------

extern "C" void kernel_launch(void* const* d_in, const int* in_sizes, int n_in,
                              void* d_out, int out_size, void* d_ws, size_t ws_size,
                              hipStream_t stream) {
  const float* x    = (const float*)d_in[0];
  const int*   ei   = (const int*)d_in[1];
  const float* Wz_c = (const float*)d_in[2];
  const float* bz_c = (const float*)d_in[3];
  // d_in[4] Wr_c, d_in[5] br_c: dead (H==0 => H*R==0)
  const float* Wh_c = (const float*)d_in[6];
  const float* bh_c = (const float*)d_in[7];
  const float* Wz_l = (const float*)d_in[8];
  const float* bz_l = (const float*)d_in[9];
  // d_in[10] Wr_l, d_in[11] br_l: dead
  const float* Wh_l = (const float*)d_in[12];
  const float* bh_l = (const float*)d_in[13];
  const float* Wlin = (const float*)d_in[14];
  const float* blin = (const float*)d_in[15];
  float* out = (float*)d_out;
  float* ws  = (float*)d_ws;

  const int* src = ei;
  const int* dst = ei + EE;

  // workspace layout (floats); all offsets multiple of 4 -> 16B aligned
  float* deg   = ws;             // N
  float* dinv  = ws + NN;        // N
  float* Hz    = ws + 2  * NN;   // 32N
  float* Hh    = ws + 34 * NN;   // 32N
  float* Az    = ws + 66 * NN;   // 32N
  float* Ah    = ws + 98 * NN;   // 32N   (total 130N floats = 26 MB)
  float* Hrelu = Hz;             // reuse: Hz dead after k_edge

  // zero accumulators every call (graph replay: no one re-clears for us)
  k_zero<<<(NN + 255) / 256, 256, 0, stream>>>(deg, NN);
  k_zero<<<(64 * NN + 255) / 256, 256, 0, stream>>>(Az, 64 * NN);

  k_degree<<<(EE + 255) / 256, 256, 0, stream>>>(dst, deg);
  k_dinv<<<(NN + 255) / 256, 256, 0, stream>>>(deg, dinv);

  k_feat<<<(NT + 7) / 8, 256, 0, stream>>>(x, Wz_c, Wh_c, Hz, Hh);

  k_edge<<<(EE * 8 + 255) / 256, 256, 0, stream>>>(src, dst, dinv, Hz, Hh, Az, Ah);

  k_gate<<<(NT + 7) / 8, 256, 0, stream>>>(Az, Ah, bz_c, bh_c, Wz_l, Wh_l, bz_l, bh_l, Hrelu);

  k_out<<<(NN + 255) / 256, 256, 0, stream>>>(Hrelu, Wlin, blin, out);
}